// Model_1778116460935
// MI455X (gfx1250) — compile-verified
//
#include <hip/hip_runtime.h>

// ---------------------------------------------------------------------------
// TransformerConv (PyG-style) for MI455X / gfx1250.
// Dense parts use V_WMMA_F32_16X16X4_F32 (full fp32, matches the fp32
// reference; GEMMs are a tiny share of runtime — the edge phase is
// L2-resident random traffic, so no precision downgrade is warranted).
// Edge phase uses native u32 atomic-max (monotonic float keys) for the
// segment max, and *native* GLOBAL_ATOMIC_ADD_F32 via inline asm for the
// segment sum / scatter aggregation (avoids HIP's possible CAS-loop
// expansion of float atomicAdd — ~102M atomics in the aggregate phase).
// ---------------------------------------------------------------------------

#define N_NODES 50000
#define EDGES   800000
#define F_IN    128
#define HEADS   4
#define CDIM    32
#define HC      128   // HEADS * CDIM
#define OUTD    64

typedef float v2f __attribute__((ext_vector_type(2)));
typedef float v8f __attribute__((ext_vector_type(8)));

// Native f32 atomic add (no return): GLOBAL_ATOMIC_ADD_F32, device scope.
__device__ __forceinline__ void atomic_add_f32_native(float* p, float v) {
  asm volatile("global_atomic_add_f32 %0, %1, off" : : "v"(p), "v"(v) : "memory");
}

// ---------------------------------------------------------------------------
// fp32 GEMM via WMMA 16x16x4.
//   A: [M,128] row-major, B: [128,NCOLS] row-major, C: [M,NCOLS].
//   One wave computes a 16x64 tile (4 accumulators) so the A fragment is
//   reused 4x per K-step.
// Fragment layouts (ISA 7.12.2, wave32):
//   A 16x4 : lane -> M = lane%16, VGPR j -> K = 2*(lane/16) + j
//   B 4x16 : VGPR j -> K = j + 2*(lane/16), N = lane%16
//   C/D    : VGPR g -> M = g + 8*(lane/16), N = lane%16
// ---------------------------------------------------------------------------
template <bool RELU_A, int NCOLS>
__global__ __launch_bounds__(256) void gemm_f32_wmma(
    const float* __restrict__ A, const float* __restrict__ B,
    const float* __restrict__ bias, float* __restrict__ C, int M) {
  constexpr int K   = 128;
  constexpr int NTQ = NCOLS / 64;  // 16x64 tiles per row-stripe
  const int lane  = threadIdx.x & 31;
  const int wid   = blockIdx.x * (blockDim.x >> 5) + (threadIdx.x >> 5);
  const int mtile = wid / NTQ;
  const int nq    = wid - mtile * NTQ;
  if (mtile * 16 >= M) return;

  const int mrow  = mtile * 16 + (lane & 15);
  const int khalf = (lane >> 4) << 1;        // 0 or 2
  const int ncol  = nq * 64 + (lane & 15);

  const float* Arow = A + (size_t)mrow * K;

  v8f acc0 = {}, acc1 = {}, acc2 = {}, acc3 = {};

#pragma unroll 4
  for (int k0 = 0; k0 < K; k0 += 4) {
    float a0 = Arow[k0 + khalf];
    float a1 = Arow[k0 + khalf + 1];
    if (RELU_A) { a0 = fmaxf(a0, 0.0f); a1 = fmaxf(a1, 0.0f); }
    v2f a; a.x = a0; a.y = a1;

    const float* B0 = B + (size_t)(k0 + khalf) * NCOLS + ncol;  // K = khalf
    const float* B1 = B0 + NCOLS;                               // K = khalf+1
    v2f b0; b0.x = B0[0];  b0.y = B1[0];
    v2f b1; b1.x = B0[16]; b1.y = B1[16];
    v2f b2; b2.x = B0[32]; b2.y = B1[32];
    v2f b3; b3.x = B0[48]; b3.y = B1[48];

    acc0 = __builtin_amdgcn_wmma_f32_16x16x4_f32(false, a, false, b0, (short)0, acc0, false, false);
    acc1 = __builtin_amdgcn_wmma_f32_16x16x4_f32(false, a, false, b1, (short)0, acc1, false, false);
    acc2 = __builtin_amdgcn_wmma_f32_16x16x4_f32(false, a, false, b2, (short)0, acc2, false, false);
    acc3 = __builtin_amdgcn_wmma_f32_16x16x4_f32(false, a, false, b3, (short)0, acc3, false, false);
  }

  const float bi0 = bias[ncol];
  const float bi1 = bias[ncol + 16];
  const float bi2 = bias[ncol + 32];
  const float bi3 = bias[ncol + 48];
  const int rbase = mtile * 16 + ((lane >> 4) << 3);
#pragma unroll
  for (int g = 0; g < 8; ++g) {
    float* Cr = C + (size_t)(rbase + g) * NCOLS + ncol;
    Cr[0]  = acc0[g] + bi0;
    Cr[16] = acc1[g] + bi1;
    Cr[32] = acc2[g] + bi2;
    Cr[48] = acc3[g] + bi3;
  }
}

// --------------------------- segment-softmax helpers -----------------------
// Monotonic uint key for float so segment-max is a single native
// GLOBAL_ATOMIC_MAX_U32 (no CAS loop). Key 0 < key(-inf), used as init.
__device__ __forceinline__ unsigned fkey(float f) {
  int i = __float_as_int(f);
  return (i >= 0) ? ((unsigned)i | 0x80000000u) : ~(unsigned)i;
}
__device__ __forceinline__ float fkey_dec(unsigned k) {
  return __int_as_float((k & 0x80000000u) ? (int)(k & 0x7fffffffu) : (int)~k);
}

__global__ void init_seg(unsigned* __restrict__ amaxk, float* __restrict__ denom, int n) {
  int i = blockIdx.x * blockDim.x + threadIdx.x;
  if (i < n) { amaxk[i] = 0u; denom[i] = 0.0f; }
}

// One wave per edge: lane = channel within head; 4 head iterations.
// alpha[e,h] = dot(q[dst,h,:], k[src,h,:]) / sqrt(32); atomic segment max.
__global__ __launch_bounds__(256) void edge_logits(
    const float* __restrict__ q, const float* __restrict__ kk,
    const int* __restrict__ ei, float* __restrict__ alpha,
    unsigned* __restrict__ amaxk) {
  const int e = blockIdx.x * (blockDim.x >> 5) + (threadIdx.x >> 5);
  if (e >= EDGES) return;
  const int lane = threadIdx.x & 31;
  const int src  = ei[e];
  const int dst  = ei[EDGES + e];
  const float* qr = q  + (size_t)dst * HC;
  const float* kr = kk + (size_t)src * HC;
#pragma unroll
  for (int h = 0; h < HEADS; ++h) {
    float p = qr[h * CDIM + lane] * kr[h * CDIM + lane];
#pragma unroll
    for (int off = 16; off > 0; off >>= 1) p += __shfl_xor(p, off, 32);
    if (lane == 0) {
      const float a = p * 0.17677669529663688f;  // 1/sqrt(32)
      alpha[(size_t)e * HEADS + h] = a;
      atomicMax(amaxk + (size_t)dst * HEADS + h, fkey(a));
    }
  }
}

// ex = exp(alpha - amax[dst]); denom[dst] += ex  (one thread per edge*head)
__global__ void edge_exp(const int* __restrict__ ei, float* __restrict__ alpha,
                         const unsigned* __restrict__ amaxk,
                         float* __restrict__ denom) {
  const long long i = (long long)blockIdx.x * blockDim.x + threadIdx.x;
  if (i >= (long long)EDGES * HEADS) return;
  const int e   = (int)(i >> 2);
  const int h   = (int)(i & 3);
  const int dst = ei[EDGES + e];
  const float m  = fkey_dec(amaxk[(size_t)dst * HEADS + h]);
  const float ex = __expf(alpha[i] - m);
  alpha[i] = ex;
  atomic_add_f32_native(denom + (size_t)dst * HEADS + h, ex);
}

// h[dst, h*32+c] += (ex / (denom[dst]+1e-16)) * v[src, h*32+c]
__global__ __launch_bounds__(256) void edge_aggregate(
    const int* __restrict__ ei, const float* __restrict__ ex,
    const float* __restrict__ denom, const float* __restrict__ v,
    float* __restrict__ hout) {
  const int e = blockIdx.x * (blockDim.x >> 5) + (threadIdx.x >> 5);
  if (e >= EDGES) return;
  const int lane = threadIdx.x & 31;
  const int src  = ei[e];
  const int dst  = ei[EDGES + e];
#pragma unroll
  for (int h = 0; h < HEADS; ++h) {
    const float attn = ex[(size_t)e * HEADS + h] /
                       (denom[(size_t)dst * HEADS + h] + 1e-16f);
    const float val = attn * v[(size_t)src * HC + h * CDIM + lane];
    atomic_add_f32_native(hout + (size_t)dst * HC + h * CDIM + lane, val);
  }
}

// ---------------------------------------------------------------------------
extern "C" void kernel_launch(void* const* d_in, const int* in_sizes, int n_in,
                              void* d_out, int out_size, void* d_ws, size_t ws_size,
                              hipStream_t stream) {
  const float* x  = (const float*)d_in[0];
  const int*   ei = (const int*)d_in[1];
  // d_in[2] = edge_weight (unused by the reference)
  const float* Wq = (const float*)d_in[3];  const float* bq = (const float*)d_in[4];
  const float* Wk = (const float*)d_in[5];  const float* bk = (const float*)d_in[6];
  const float* Wv = (const float*)d_in[7];  const float* bv = (const float*)d_in[8];
  const float* Ws = (const float*)d_in[9];  const float* bs = (const float*)d_in[10];
  const float* Wl = (const float*)d_in[11]; const float* bl = (const float*)d_in[12];

  float* y = (float*)d_out;                      // [N, 64]
  float* h = y + (size_t)N_NODES * OUTD;         // [N, 128] (second tuple elem)

  // workspace layout (~91 MB)
  float* q  = (float*)d_ws;                              // [N, 128]
  float* kk = q  + (size_t)N_NODES * HC;                 // [N, 128]
  float* v  = kk + (size_t)N_NODES * HC;                 // [N, 128]
  float* al = v  + (size_t)N_NODES * HC;                 // [E, 4] alpha -> ex
  unsigned* amaxk = (unsigned*)(al + (size_t)EDGES * HEADS);   // [N, 4]
  float* denom    = (float*)(amaxk + (size_t)N_NODES * HEADS); // [N, 4]

  {  // init segment buffers
    const int n = N_NODES * HEADS;
    init_seg<<<(n + 255) / 256, 256, 0, stream>>>(amaxk, denom, n);
  }
  {  // projections: q, k, v -> ws; skip -> h region of d_out (accumulated into)
    const int waves  = (N_NODES / 16) * (HC / 64);   // 6250
    const int blocks = (waves + 7) / 8;
    gemm_f32_wmma<false, HC><<<blocks, 256, 0, stream>>>(x, Wq, bq, q,  N_NODES);
    gemm_f32_wmma<false, HC><<<blocks, 256, 0, stream>>>(x, Wk, bk, kk, N_NODES);
    gemm_f32_wmma<false, HC><<<blocks, 256, 0, stream>>>(x, Wv, bv, v,  N_NODES);
    gemm_f32_wmma<false, HC><<<blocks, 256, 0, stream>>>(x, Ws, bs, h,  N_NODES);
  }
  {  // attention logits + segment max
    const int blocks = (EDGES + 7) / 8;
    edge_logits<<<blocks, 256, 0, stream>>>(q, kk, ei, al, amaxk);
  }
  {  // exp + segment sum
    const long long n = (long long)EDGES * HEADS;
    edge_exp<<<(int)((n + 255) / 256), 256, 0, stream>>>(ei, al, amaxk, denom);
  }
  {  // weighted scatter-aggregate into h
    const int blocks = (EDGES + 7) / 8;
    edge_aggregate<<<blocks, 256, 0, stream>>>(ei, al, denom, v, h);
  }
  {  // y = relu(h) @ Wlin + blin
    const int waves  = (N_NODES / 16) * (OUTD / 64); // 3125
    const int blocks = (waves + 7) / 8;
    gemm_f32_wmma<true, OUTD><<<blocks, 256, 0, stream>>>(h, Wl, bl, y, N_NODES);
  }
}